// PointnetSAModuleMSG_22505628631265
// MI455X (gfx1250) — compile-verified
//
#include <hip/hip_runtime.h>
#include <stdint.h>

typedef __attribute__((ext_vector_type(2))) float v2f;
typedef __attribute__((ext_vector_type(8))) float v8f;

#define B_   16
#define N_   4096
#define CF_  64
#define S_   1024
#define EPS_ 1e-5f
#define NCHUNK_ 256

// ---------------------------------------------------------------- FPS
__global__ __launch_bounds__(256) void fps_kernel(const float* __restrict__ xyz,
                                                  int* __restrict__ fpsIdx) {
  __shared__ float sx[N_ * 3];
  __shared__ float rv[256];
  __shared__ int   ri[256];
  const int b = blockIdx.x, tid = threadIdx.x;
  for (int i = tid; i < N_ * 3; i += 256) sx[i] = xyz[(size_t)b * N_ * 3 + i];
  float dist[16];
#pragma unroll
  for (int t = 0; t < 16; ++t) dist[t] = 1e10f;
  __syncthreads();
  int far = 0;
  for (int j = 0; j < S_; ++j) {
    if (tid == 0) fpsIdx[b * S_ + j] = far;
    const float cx = sx[far * 3 + 0], cy = sx[far * 3 + 1], cz = sx[far * 3 + 2];
    float bv = -1.0f; int bi = 0;
#pragma unroll
    for (int t = 0; t < 16; ++t) {
      const int p = t * 256 + tid;            // lane-stride-3 LDS pattern: conflict-free
      const float dx = sx[p * 3 + 0] - cx;
      const float dy = sx[p * 3 + 1] - cy;
      const float dz = sx[p * 3 + 2] - cz;
      const float d  = dx * dx + dy * dy + dz * dz;
      const float nd = fminf(dist[t], d);
      dist[t] = nd;
      if (nd > bv) { bv = nd; bi = p; }       // ascending p => first occurrence kept
    }
    rv[tid] = bv; ri[tid] = bi;
    __syncthreads();
    for (int off = 128; off > 0; off >>= 1) {
      if (tid < off) {
        const float ov = rv[tid + off]; const int oi = ri[tid + off];
        if (ov > rv[tid] || (ov == rv[tid] && oi < ri[tid])) { rv[tid] = ov; ri[tid] = oi; }
      }
      __syncthreads();
    }
    far = ri[0];
    __syncthreads();
  }
}

// ------------------------------------------------------- gather new_xyz
__global__ void gather_newxyz_kernel(const float* __restrict__ xyz,
                                     const int* __restrict__ fpsIdx,
                                     float* __restrict__ out) {
  const int e = blockIdx.x * 256 + threadIdx.x;
  if (e >= B_ * S_ * 3) return;
  const int d = e % 3;
  const int bs = e / 3;
  const int s = bs % S_, b = bs / S_;
  const int n = fpsIdx[b * S_ + s];
  out[e] = xyz[((size_t)b * N_ + n) * 3 + d];
}

// ------------------------------------------------------- dual-scale ball query
__global__ __launch_bounds__(256) void ballquery_kernel(const float* __restrict__ xyz,
                                                        const float* __restrict__ newxyz,
                                                        int* __restrict__ idx0,
                                                        int* __restrict__ idx1) {
  __shared__ float tile[512 * 3];
  const int tid = threadIdx.x;
  const int bs  = blockIdx.x * 256 + tid;     // one center per thread
  const int b   = blockIdx.x >> 2;            // S/256 = 4 blocks per batch
  const float cx = newxyz[bs * 3 + 0], cy = newxyz[bs * 3 + 1], cz = newxyz[bs * 3 + 2];
  const float r0 = 0.25f * 0.25f, r1 = 0.5f * 0.5f;
  int cnt0 = 0, cnt1 = 0;
  const int64_t base0 = (int64_t)bs * 32, base1 = (int64_t)bs * 64;
  for (int n0 = 0; n0 < N_; n0 += 512) {
    for (int i = tid; i < 512 * 3; i += 256)
      tile[i] = xyz[((size_t)b * N_ + n0) * 3 + i];
    __syncthreads();
    if (cnt0 < 32 || cnt1 < 64) {
      for (int t = 0; t < 512; ++t) {
        const float dx = tile[t * 3 + 0] - cx;
        const float dy = tile[t * 3 + 1] - cy;
        const float dz = tile[t * 3 + 2] - cz;
        const float d2 = dx * dx + dy * dy + dz * dz;
        if (d2 < r1) {
          const int n = n0 + t;
          if (d2 < r0 && cnt0 < 32) idx0[base0 + cnt0++] = n;
          if (cnt1 < 64)            idx1[base1 + cnt1++] = n;
        }
      }
    }
    __syncthreads();
  }
  const int f0 = idx0[base0];                 // center itself guarantees >=1 hit
  for (int k = cnt0; k < 32; ++k) idx0[base0 + k] = f0;
  const int f1 = idx1[base1];
  for (int k = cnt1; k < 64; ++k) idx1[base1 + k] = f1;
}

// ------------------------------------------------------- build padded GEMM input (M x 68)
__global__ void build_input_kernel(const float* __restrict__ xyz,
                                   const float* __restrict__ feat,
                                   const float* __restrict__ newxyz,
                                   const int* __restrict__ idx, int K,
                                   float* __restrict__ IN, int64_t M) {
  const int64_t e = (int64_t)blockIdx.x * 256 + threadIdx.x;
  if (e >= M * 68) return;
  const int64_t m = e / 68;
  const int c = (int)(e % 68);
  const int n = idx[m];
  const int64_t bs = m / K;
  const int64_t b = bs / S_;
  float v;
  if (c < 3)       v = xyz[((int64_t)b * N_ + n) * 3 + c] - newxyz[bs * 3 + c];
  else if (c < 67) v = feat[((int64_t)b * N_ + n) * CF_ + (c - 3)];
  else             v = 0.f;
  IN[e] = v;
}

// ------------------------------------------------------- pack W into WMMA B-fragment order
// Bpack layout (float2 units): [nblk][kstep][tg(0..3)][lane(0..31)]
//   col = nblk*64 + tg*16 + (lane&15), kk = kstep*4 + (lane>>4)*2, pair = W[col][kk..kk+1]
__global__ void pack_weights_kernel(const float* __restrict__ W, int cinW, int cinPad,
                                    int cout, float2* __restrict__ Bpack) {
  const int total = (cout / 64) * cinPad * 32;  // pairs
  const int e = blockIdx.x * 256 + threadIdx.x;
  if (e >= total) return;
  const int perPanel = cinPad * 32;
  const int nblk = e / perPanel;
  const int r = e % perPanel;
  const int kstep = r >> 7;          // / 128
  const int tg = (r >> 5) & 3;
  const int lane = r & 31;
  const int col = nblk * 64 + tg * 16 + (lane & 15);
  const int kk = kstep * 4 + (lane >> 4) * 2;
  float2 p;
  p.x = (kk     < cinW) ? W[(int64_t)col * cinW + kk]     : 0.f;
  p.y = (kk + 1 < cinW) ? W[(int64_t)col * cinW + kk + 1] : 0.f;
  Bpack[e] = p;
}

// ------------------------------------------------------- fp32 WMMA GEMM
// per wave: 16(M) x 64(N), K stepped by 4 via v_wmma_f32_16x16x4_f32.
// B panel staged to LDS with global_load_async_to_lds_b128 (CDNA5 async path).
__global__ __launch_bounds__(256) void wmma_gemm_kernel(const float* __restrict__ A, int lda,
                                                        const float2* __restrict__ Bpack,
                                                        const float* __restrict__ bias,
                                                        float* __restrict__ Cmat, int cout) {
  __shared__ float sB[128 * 64];  // up to 32 KB: cinPad(<=128) * 64 cols
  const int lane = threadIdx.x & 31;
  const int w    = threadIdx.x >> 5;
  const int64_t mBase = (int64_t)blockIdx.x * 128 + w * 16;
  const int nBase = blockIdx.y * 64;
  const int l16  = lane & 15;
  const int kOff = (lane >> 4) * 2;

  // ---- async-stage this block's B panel (lda*64 floats) into LDS ----
  {
    const char* gsrc = (const char*)(Bpack + (size_t)blockIdx.y * (lda * 32));
    const unsigned sbase = (unsigned)(uintptr_t)sB;       // low 32 bits = LDS byte offset
    const int lines = (lda * 64 * 4) >> 4;                // 16-byte lines
    for (int i = threadIdx.x; i < lines; i += 256) {
      const unsigned ldsAddr = sbase + (unsigned)(i << 4);
      const uint64_t ga = (uint64_t)(uintptr_t)(gsrc + ((size_t)i << 4));
      asm volatile("global_load_async_to_lds_b128 %0, %1, off"
                   :: "v"(ldsAddr), "v"(ga) : "memory");
    }
    asm volatile("s_wait_asynccnt 0x0" ::: "memory");
  }
  __syncthreads();

  const v8f zero = {0.f, 0.f, 0.f, 0.f, 0.f, 0.f, 0.f, 0.f};
  v8f acc[4];
#pragma unroll
  for (int t = 0; t < 4; ++t) acc[t] = zero;

  const float* Arow = A + (mBase + l16) * lda;
  const v2f* sB2 = (const v2f*)sB;
  const int ksteps = lda >> 2;
  for (int ks = 0; ks < ksteps; ++ks) {
    __builtin_prefetch(Arow + ks * 4 + 64, 0, 1);         // -> global_prefetch_b8
    const v2f a = *(const v2f*)(Arow + ks * 4 + kOff);    // one global_load_b64
#pragma unroll
    for (int t = 0; t < 4; ++t) {
      const v2f bf = sB2[ks * 128 + t * 32 + lane];       // ds_load_b64, conflict-free
      acc[t] = __builtin_amdgcn_wmma_f32_16x16x4_f32(false, a, false, bf,
                                                     (short)0, acc[t], false, false);
    }
  }

#pragma unroll
  for (int t = 0; t < 4; ++t) {
    const int col = nBase + t * 16 + l16;
    const float bb = bias[col];
    const int64_t r0 = mBase + 8 * (lane >> 4);
#pragma unroll
    for (int r = 0; r < 8; ++r)
      Cmat[(r0 + r) * cout + col] = acc[t][r] + bb;
  }
}

// ------------------------------------------------------- BN stats, deterministic 2-stage
__global__ __launch_bounds__(256) void stats1_kernel(const float* __restrict__ X, int64_t M,
                                                     int cout, float* __restrict__ partial) {
  const int c = blockIdx.x * 256 + threadIdx.x;
  if (c >= cout) return;
  const int64_t rowsPer = M / gridDim.y;
  const int64_t r0 = (int64_t)blockIdx.y * rowsPer;
  float s = 0.f, sq = 0.f;
  for (int64_t r = r0; r < r0 + rowsPer; ++r) {
    const float v = X[r * cout + c];
    s += v; sq += v * v;
  }
  partial[(int64_t)blockIdx.y * 2 * cout + c] = s;
  partial[(int64_t)blockIdx.y * 2 * cout + cout + c] = sq;
}

__global__ void stats2_kernel(const float* __restrict__ partial, int nchunk, int64_t M,
                              int cout, float* __restrict__ meanr, float* __restrict__ rstd) {
  const int c = blockIdx.x * 256 + threadIdx.x;
  if (c >= cout) return;
  float s = 0.f, sq = 0.f;
  for (int i = 0; i < nchunk; ++i) {
    s  += partial[(int64_t)i * 2 * cout + c];
    sq += partial[(int64_t)i * 2 * cout + cout + c];
  }
  const float m = s / (float)M;
  const float v = sq / (float)M - m * m;
  meanr[c] = m;
  rstd[c] = rsqrtf(v + EPS_);
}

// ------------------------------------------------------- BN + ReLU in place
__global__ void bn_relu_kernel(float* __restrict__ X, int64_t total, int cout,
                               const float* __restrict__ meanr, const float* __restrict__ rstd,
                               const float* __restrict__ g, const float* __restrict__ be) {
  const int64_t e = (int64_t)blockIdx.x * 256 + threadIdx.x;
  if (e >= total) return;
  const int c = (int)(e % cout);
  const float y = g[c] * (X[e] - meanr[c]) * rstd[c] + be[c];
  X[e] = fmaxf(y, 0.f);
}

// ------------------------------------------------------- fused BN+ReLU+max over (K,S)
__global__ __launch_bounds__(256) void final_max_kernel(const float* __restrict__ X, int K,
                                                        int cout,
                                                        const float* __restrict__ meanr,
                                                        const float* __restrict__ rstd,
                                                        const float* __restrict__ g,
                                                        const float* __restrict__ be,
                                                        int* __restrict__ outBits,
                                                        int outStride, int outOff) {
  const int c = blockIdx.y * 256 + threadIdx.x;
  if (c >= cout) return;
  const int b = blockIdx.x;
  const int64_t rowsPerB = (int64_t)S_ * K;
  const int64_t rowsPer = rowsPerB / gridDim.z;
  const int64_t r0 = (int64_t)b * rowsPerB + (int64_t)blockIdx.z * rowsPer;
  const float mc = meanr[c], rc = rstd[c], gc = g[c], bc = be[c];
  float gm = -1.f;
  for (int64_t r = r0; r < r0 + rowsPer; ++r) {
    const float y = gc * (X[r * cout + c] - mc) * rc + bc;
    gm = fmaxf(gm, y);
  }
  gm = fmaxf(gm, 0.f);  // ReLU; result >= 0 so int-bit atomicMax is order-preserving
  atomicMax(&outBits[b * outStride + outOff + c], __float_as_int(gm));
}

// =================================================================== launcher
static inline unsigned cdiv(int64_t a, int64_t b) { return (unsigned)((a + b - 1) / b); }

extern "C" void kernel_launch(void* const* d_in, const int* in_sizes, int n_in,
                              void* d_out, int out_size, void* d_ws, size_t ws_size,
                              hipStream_t stream) {
  const float* xyz  = (const float*)d_in[0];
  const float* feat = (const float*)d_in[1];
  // params flattened: scale-major, layer-major, then (W, b, g, be)
  auto P = [&](int scale, int layer, int which) -> const float* {
    return (const float*)d_in[2 + scale * 12 + layer * 4 + which];
  };

  const int64_t M1 = (int64_t)B_ * S_ * 64;  // largest M (scale 1)

  char* ws = (char*)d_ws;
  size_t off = 0;
  auto alloc = [&](size_t bytes) -> char* {
    char* p = ws + off;
    off += (bytes + 255) & ~(size_t)255;
    return p;
  };
  int*   fpsIdx  = (int*)alloc((size_t)B_ * S_ * 4);
  int*   idx0    = (int*)alloc((size_t)B_ * S_ * 32 * 4);
  int*   idx1    = (int*)alloc((size_t)B_ * S_ * 64 * 4);
  float* IN      = (float*)alloc((size_t)M1 * 68 * 4);
  float* T1      = (float*)alloc((size_t)M1 * 128 * 4);
  float* T2      = (float*)alloc((size_t)M1 * 128 * 4);
  float* T3      = (float*)alloc((size_t)M1 * 512 * 4);
  float* partial = (float*)alloc((size_t)NCHUNK_ * 2 * 512 * 4);
  float* meanr   = (float*)alloc(512 * 4);
  float* rstd    = (float*)alloc(512 * 4);

  float* newxyz  = (float*)d_out;                       // (B,S,3) = 49152 floats
  int*   outBits = (int*)((float*)d_out + B_ * S_ * 3); // (B,768) feature maxes

  const int Ks[2]       = {32, 64};
  const int cins[2][3]  = {{67, 64, 128}, {67, 128, 128}};   // true cin
  const int cinp[2][3]  = {{68, 64, 128}, {68, 128, 128}};   // padded cin (mult of 4)
  const int couts[2][3] = {{64, 128, 256}, {128, 128, 512}};
  const int outOffs[2]  = {0, 256};
  int* idxs[2] = {idx0, idx1};

  // packed-weight arena (fragment-ordered, zero-padded)
  float2* Bp[2][3];
  for (int sc = 0; sc < 2; ++sc)
    for (int l = 0; l < 3; ++l)
      Bp[sc][l] = (float2*)alloc((size_t)cinp[sc][l] * couts[sc][l] * 4);

  // zero the feature-max region (bits 0 == 0.0f), capture-legal
  hipMemsetAsync(outBits, 0, (size_t)B_ * 768 * 4, stream);

  // pack all weights once (tiny)
  for (int sc = 0; sc < 2; ++sc)
    for (int l = 0; l < 3; ++l) {
      const int pairs = (couts[sc][l] / 64) * cinp[sc][l] * 32;
      pack_weights_kernel<<<cdiv(pairs, 256), 256, 0, stream>>>(
          P(sc, l, 0), cins[sc][l], cinp[sc][l], couts[sc][l], Bp[sc][l]);
    }

  fps_kernel<<<B_, 256, 0, stream>>>(xyz, fpsIdx);
  gather_newxyz_kernel<<<cdiv((int64_t)B_ * S_ * 3, 256), 256, 0, stream>>>(xyz, fpsIdx, newxyz);
  ballquery_kernel<<<B_ * (S_ / 256), 256, 0, stream>>>(xyz, newxyz, idx0, idx1);

  for (int sc = 0; sc < 2; ++sc) {
    const int K = Ks[sc];
    const int64_t M = (int64_t)B_ * S_ * K;
    const int co0 = couts[sc][0], co1 = couts[sc][1], co2 = couts[sc][2];

    build_input_kernel<<<cdiv(M * 68, 256), 256, 0, stream>>>(xyz, feat, newxyz,
                                                              idxs[sc], K, IN, M);

    // layer 1: IN (M x 68) -> T1 (M x co0)
    wmma_gemm_kernel<<<dim3((unsigned)(M / 128), co0 / 64), 256, 0, stream>>>(
        IN, 68, Bp[sc][0], P(sc, 0, 1), T1, co0);
    stats1_kernel<<<dim3(cdiv(co0, 256), NCHUNK_), 256, 0, stream>>>(T1, M, co0, partial);
    stats2_kernel<<<cdiv(co0, 256), 256, 0, stream>>>(partial, NCHUNK_, M, co0, meanr, rstd);
    bn_relu_kernel<<<cdiv(M * co0, 256), 256, 0, stream>>>(T1, M * co0, co0, meanr, rstd,
                                                           P(sc, 0, 2), P(sc, 0, 3));

    // layer 2: T1 (M x co0) -> T2 (M x co1)
    wmma_gemm_kernel<<<dim3((unsigned)(M / 128), co1 / 64), 256, 0, stream>>>(
        T1, co0, Bp[sc][1], P(sc, 1, 1), T2, co1);
    stats1_kernel<<<dim3(cdiv(co1, 256), NCHUNK_), 256, 0, stream>>>(T2, M, co1, partial);
    stats2_kernel<<<cdiv(co1, 256), 256, 0, stream>>>(partial, NCHUNK_, M, co1, meanr, rstd);
    bn_relu_kernel<<<cdiv(M * co1, 256), 256, 0, stream>>>(T2, M * co1, co1, meanr, rstd,
                                                           P(sc, 1, 2), P(sc, 1, 3));

    // layer 3: T2 (M x co1) -> T3 (M x co2), BN stats, fused BN+ReLU+max
    wmma_gemm_kernel<<<dim3((unsigned)(M / 128), co2 / 64), 256, 0, stream>>>(
        T2, co1, Bp[sc][2], P(sc, 2, 1), T3, co2);
    stats1_kernel<<<dim3(cdiv(co2, 256), NCHUNK_), 256, 0, stream>>>(T3, M, co2, partial);
    stats2_kernel<<<cdiv(co2, 256), 256, 0, stream>>>(partial, NCHUNK_, M, co2, meanr, rstd);
    final_max_kernel<<<dim3(B_, cdiv(co2, 256), 32), 256, 0, stream>>>(
        T3, K, co2, meanr, rstd, P(sc, 2, 2), P(sc, 2, 3), outBits, 768, outOffs[sc]);
  }
}